// SymbolicMathExpert_17197049053709
// MI455X (gfx1250) — compile-verified
//
#include <hip/hip_runtime.h>
#include <math.h>

#define CLAMPV  1.0e6f
#define LN_EPS  1e-5f

constexpr int D_DIM    = 1024;
constexpr int C_DIM    = 10;
constexpr int NPAD     = 16;
constexpr int TOK_BLK  = 16;              // tokens per workgroup (= waves)
constexpr int LANES    = 32;              // wave32
constexpr int EPL      = D_DIM / LANES;   // 32 elements per lane
constexpr int VEC      = 4;
constexpr int NV       = EPL / VEC;
constexpr int XSTRIDE  = D_DIM + 4;

typedef __attribute__((ext_vector_type(2))) float v2f;
typedef __attribute__((ext_vector_type(8))) float v8f;

__device__ __forceinline__ float wave_sum32(float v) {
#pragma unroll
    for (int off = 16; off >= 1; off >>= 1)
        v += __shfl_xor(v, off, 32);
    return v;
}

__device__ __forceinline__ float clampv(float v) {      // v_med3_num_f32
    return fminf(fmaxf(v, -CLAMPV), CLAMPV);
}

__device__ __forceinline__ float fast_exp(float x) {    // e^x via v_exp (exp2)
    return __builtin_amdgcn_exp2f(x * 1.4426950408889634f);
}
__device__ __forceinline__ float fast_tanh(float x) {
#if __has_builtin(__builtin_amdgcn_tanhf)
    return __builtin_amdgcn_tanhf(x);
#else
    const float e = __builtin_amdgcn_exp2f(-2.8853900817779268f * fabsf(x));
    const float t = (1.0f - e) * __builtin_amdgcn_rcpf(1.0f + e);
    return copysignf(t, x);
#endif
}

__global__ void pad_gate_kernel(const float* __restrict__ gw, float* __restrict__ gwpad) {
    const int idx = blockIdx.x * blockDim.x + threadIdx.x;
    gwpad[idx] = ((idx >> 10) < C_DIM) ? gw[idx] : 0.0f;
}

__global__ __launch_bounds__(TOK_BLK * LANES, 1)
void symexp_fused_kernel(const float* __restrict__ x,
                         const float* __restrict__ gamma,
                         const float* __restrict__ beta,
                         const float* __restrict__ lis,
                         const float* __restrict__ los,
                         const float* __restrict__ gwpad,
                         const float* __restrict__ gb,
                         float* __restrict__ outw,
                         float* __restrict__ outg)
{
    __shared__ float xtile[TOK_BLK * XSTRIDE];
    __shared__ float logits[16 * 16];

    const int tid  = threadIdx.x;
    const int w    = tid >> 5;
    const int lane = tid & 31;
    const int tok  = blockIdx.x * TOK_BLK + w;

    // ---- stage raw x (b128) -> regs + LDS ----
    float xv[EPL];
    const float* xrow = x + (size_t)tok * D_DIM;
#pragma unroll
    for (int i = 0; i < NV; ++i) {
        const int d = i * (LANES * VEC) + lane * VEC;
        const float4 v4 = *(const float4*)(xrow + d);
        xv[i * VEC + 0] = v4.x; xv[i * VEC + 1] = v4.y;
        xv[i * VEC + 2] = v4.z; xv[i * VEC + 3] = v4.w;
        *(float4*)(&xtile[w * XSTRIDE + d]) = v4;
    }
    if (tid < 256) logits[tid] = 0.0f;
    __syncthreads();

    // ---- input LayerNorm over D ----
    float s = 0.0f, s2 = 0.0f;
#pragma unroll
    for (int i = 0; i < EPL; ++i) { s += xv[i]; s2 = fmaf(xv[i], xv[i], s2); }
    s  = wave_sum32(s);
    s2 = wave_sum32(s2);
    const float mu   = s * (1.0f / D_DIM);
    const float var  = s2 * (1.0f / D_DIM) - mu * mu;
    const float rstd = __frsqrt_rn(var + LN_EPS);
#pragma unroll
    for (int i = 0; i < NV; ++i) {
        const int d = i * (LANES * VEC) + lane * VEC;
        const float4 g4 = *(const float4*)(gamma + d);
        const float4 b4 = *(const float4*)(beta + d);
        xv[i * VEC + 0] = (xv[i * VEC + 0] - mu) * rstd * g4.x + b4.x;
        xv[i * VEC + 1] = (xv[i * VEC + 1] - mu) * rstd * g4.y + b4.y;
        xv[i * VEC + 2] = (xv[i * VEC + 2] - mu) * rstd * g4.z + b4.z;
        xv[i * VEC + 3] = (xv[i * VEC + 3] - mu) * rstd * g4.w + b4.w;
    }

    // ---- candidate scales + folded coefficients ----
    // |x_norm| <= sqrt(D)*max|gamma|+max|beta| (~32): input clamp never binds.
    // All candidates except exp are bounded far below 1e6: output clamp kept on exp only.
    float ins[C_DIM], outs[C_DIM];
#pragma unroll
    for (int c = 0; c < C_DIM; ++c) { ins[c] = fast_exp(lis[c]); outs[c] = fast_exp(los[c]); }

    const float A  = ins[0] * outs[0];                            // c0: A*x
    const float Bq = ins[1] * ins[1] * outs[1];                   // c1: B*x^2
    const float Cc = ins[6] * ins[6] * ins[6] * outs[6];          // c6: C*x^3
    const float A2 = A * A, B2 = Bq * Bq, C2 = Cc * Cc;
    const float k2i = ins[2];                                     // v_cos is period-1: cos(2pi*(x*ins2))
    const float k3i = ins[3] * 0.15915494309189535f;              // v_sin is period-1
    const float k4i = ins[4] * 1.4426950408889634f;               // exp via exp2
    const float k7o = outs[7] * 0.6931471805599453f;              // ln = ln2 * log2
    const float k8o = __builtin_amdgcn_sqrtf(ins[8]) * outs[8];   // sqrt(|x*s|) = sqrt(s)*sqrt(|x|)
    const float v5  = clampv(outs[5]);                            // c5 constant

    // ---- joint statistics over (C, D) ----
    float js  = (float)EPL * v5;
    float js2 = (float)EPL * v5 * v5;
#pragma unroll
    for (int i = 0; i < EPL; ++i) {
        const float xn = xv[i];
        const float y  = xn * xn;
        // polynomial trio c0,c1,c6: value-sum and square-sum in Horner form
        const float t = fmaf(Cc, y, A);
        js = fmaf(t, xn, js);
        js = fmaf(Bq, y, js);
        float q = fmaf(C2, y, B2);
        q = fmaf(q, y, A2);
        js2 = fmaf(q, y, js2);
        // transcendental candidates
        float v;
        v = __builtin_amdgcn_cosf(xn * k2i) * outs[2];            js += v; js2 = fmaf(v, v, js2);
        v = __builtin_amdgcn_sinf(xn * k3i) * outs[3];            js += v; js2 = fmaf(v, v, js2);
        v = clampv(__builtin_amdgcn_exp2f(xn * k4i) * outs[4]);   js += v; js2 = fmaf(v, v, js2);
        v = __builtin_amdgcn_logf(fabsf(xn * ins[7]) + 1.0f) * k7o; js += v; js2 = fmaf(v, v, js2);
        v = __builtin_amdgcn_sqrtf(fabsf(xn)) * k8o;              js += v; js2 = fmaf(v, v, js2);
        v = fast_tanh(xn * ins[9]) * outs[9];                     js += v; js2 = fmaf(v, v, js2);
    }
    js  = wave_sum32(js);
    js2 = wave_sum32(js2);
    const float invN = 1.0f / (float)(C_DIM * D_DIM);
    const float jmu  = js * invN;
    const float jvar = js2 * invN - jmu * jmu;
    const float jrs  = __frsqrt_rn(jvar + LN_EPS);

    // ---- gate logits via fp32 WMMA over K=1024 (wave w owns K-slice of 64) ----
    {
        const int kbase = w * (D_DIM / TOK_BLK);
        const int m     = lane & 15;
        const int ksub  = (lane < 16) ? 0 : 2;

        v2f bf[16];
        const float* bbase = gwpad + (size_t)m * D_DIM + kbase + ksub;
#pragma unroll
        for (int kk = 0; kk < 16; ++kk) {
            bf[kk].x = bbase[kk * 4 + 0];
            bf[kk].y = bbase[kk * 4 + 1];
        }

        v8f acc = {};
        const float* abase = &xtile[m * XSTRIDE + kbase + ksub];
#pragma unroll
        for (int kk = 0; kk < 16; ++kk) {
            v2f a;
            a.x = abase[kk * 4 + 0];
            a.y = abase[kk * 4 + 1];
            acc = __builtin_amdgcn_wmma_f32_16x16x4_f32(
                      false, a, false, bf[kk], (short)0, acc, false, false);
        }
        const int mrow = (lane < 16) ? 0 : 8;
#pragma unroll
        for (int r = 0; r < 8; ++r)
            atomicAdd(&logits[(mrow + r) * 16 + m], acc[r]);
    }
    __syncthreads();

    // ---- softmax over candidates ----
    float gwt[C_DIM];
    float mx = -3.4e38f;
#pragma unroll
    for (int c = 0; c < C_DIM; ++c) {
        gwt[c] = logits[w * 16 + c] + gb[c];
        mx = fmaxf(mx, gwt[c]);
    }
    float se = 0.0f;
#pragma unroll
    for (int c = 0; c < C_DIM; ++c) { gwt[c] = fast_exp(gwt[c] - mx); se += gwt[c]; }
    const float rse = __builtin_amdgcn_rcpf(se);
#pragma unroll
    for (int c = 0; c < C_DIM; ++c) gwt[c] *= rse;

    if (lane < C_DIM) outg[(size_t)tok * C_DIM + lane] = gwt[lane];

    // ---- weighted sum ----
    // Post-LN clip/nan_to_num never bind (max|v-mu| <= sqrt(N)*sigma ~ 101*sigma << 1e6*sigma,
    // all values finite) => out = (sum_c gwt*v_c - jmu) * jrs, softmax weights sum to 1.
    const float njmu = -jmu * jrs;
    const float P0 = gwt[0] * A, P1 = gwt[1] * Bq, P6 = gwt[6] * Cc;
    const float w2 = gwt[2] * outs[2];
    const float w3 = gwt[3] * outs[3];
    const float w4 = gwt[4];                    // exp keeps its own clamp
    const float w7 = gwt[7] * k7o;
    const float w8 = gwt[8] * k8o;
    const float w9 = gwt[9] * outs[9];
    const float acc0 = gwt[5] * v5;

    float* orow = outw + (size_t)tok * D_DIM;
#pragma unroll
    for (int i = 0; i < NV; ++i) {
        float4 o;
        float* op = &o.x;
#pragma unroll
        for (int j = 0; j < VEC; ++j) {
            const float xn = xv[i * VEC + j];
            const float y  = xn * xn;
            float accv = acc0;
            float t = fmaf(P6, y, P0);                    // c0 + c6
            accv = fmaf(t, xn, accv);
            accv = fmaf(P1, y, accv);                     // c1
            accv = fmaf(w2, __builtin_amdgcn_cosf(xn * k2i), accv);
            accv = fmaf(w3, __builtin_amdgcn_sinf(xn * k3i), accv);
            accv = fmaf(w4, clampv(__builtin_amdgcn_exp2f(xn * k4i) * outs[4]), accv);
            accv = fmaf(w7, __builtin_amdgcn_logf(fabsf(xn * ins[7]) + 1.0f), accv);
            accv = fmaf(w8, __builtin_amdgcn_sqrtf(fabsf(xn)), accv);
            accv = fmaf(w9, fast_tanh(xn * ins[9]), accv);
            op[j] = fmaf(accv, jrs, njmu);
        }
        *(float4*)(orow + i * (LANES * VEC) + lane * VEC) = o;
    }
}

extern "C" void kernel_launch(void* const* d_in, const int* in_sizes, int n_in,
                              void* d_out, int out_size, void* d_ws, size_t ws_size,
                              hipStream_t stream) {
    const float* x     = (const float*)d_in[0];
    const float* gamma = (const float*)d_in[1];
    const float* beta  = (const float*)d_in[2];
    const float* lis   = (const float*)d_in[3];
    const float* los   = (const float*)d_in[4];
    const float* gw    = (const float*)d_in[5];
    const float* gb    = (const float*)d_in[6];

    const int tokens = in_sizes[0] / D_DIM;          // B*S = 8192
    float* outw  = (float*)d_out;
    float* outg  = (float*)d_out + (size_t)tokens * D_DIM;
    float* gwpad = (float*)d_ws;                     // 16*1024 floats = 64 KB

    pad_gate_kernel<<<(NPAD * D_DIM) / 256, 256, 0, stream>>>(gw, gwpad);

    const int blocks = tokens / TOK_BLK;             // 512
    symexp_fused_kernel<<<blocks, TOK_BLK * LANES, 0, stream>>>(
        x, gamma, beta, lis, los, gwpad, gb, outw, outg);
}